// BestBuddyLoss_2035814499037
// MI455X (gfx1250) — compile-verified
//
#include <hip/hip_runtime.h>
#include <hip/hip_bf16.h>

typedef float v2f __attribute__((ext_vector_type(2)));
typedef float v8f __attribute__((ext_vector_type(8)));

#define BB   16
#define CC   3
#define HH   144
#define WW   144
#define NN   2304   // (144/3)^2 patches
#define MM   3024   // 2304 + 576 + 144
#define DP   32     // padded patch length (27 real)
#define NT   (MM / 16)   // 189 m-tiles

__device__ __forceinline__ int iclamp(int v, int lo, int hi) {
    return v < lo ? lo : (v > hi ? hi : v);
}

// ---------------------------------------------------------------------------
// Kernel 1: build P1 rows (from x), Q rows (gt patches + bicubic ds2/ds4
// patches computed inline), and row norms. One thread per row.
// ---------------------------------------------------------------------------
__global__ void bbl_build(const float* __restrict__ x, const float* __restrict__ gt,
                          float* __restrict__ P1, float* __restrict__ Q,
                          float* __restrict__ n1, float* __restrict__ qn)
{
    int tid = blockIdx.x * blockDim.x + threadIdx.x;
    const int totalP1 = BB * NN;
    const int totalQ  = BB * MM;
    if (tid >= totalP1 + totalQ) return;

    const float cw[4] = {-0.09375f, 0.59375f, 0.59375f, -0.09375f};

    if (tid < totalP1) {
        int b = tid / NN, n = tid % NN;
        int py = n / 48, px = n % 48;
        float* row = P1 + (size_t)tid * DP;
        float nrm = 0.f;
        #pragma unroll
        for (int k = 0; k < 27; k++) {
            int c = k / 9, r = k % 9, ky = r / 3, kx = r % 3;
            float v = x[(((size_t)b * CC + c) * HH + (py * 3 + ky)) * WW + (px * 3 + kx)];
            row[k] = v; nrm += v * v;
        }
        #pragma unroll
        for (int k = 27; k < DP; k++) row[k] = 0.f;
        n1[tid] = nrm;
    } else {
        int q = tid - totalP1;
        int b = q / MM, m = q % MM;
        float* row = Q + (size_t)q * DP;
        float nrm = 0.f;
        int f, py, px;
        if (m < 2304)      { f = 1; py = m / 48;            px = m % 48; }
        else if (m < 2880) { int i = m - 2304; f = 2; py = i / 24; px = i % 24; }
        else               { int i = m - 2880; f = 4; py = i / 12; px = i % 12; }
        #pragma unroll 1
        for (int k = 0; k < 27; k++) {
            int c = k / 9, r = k % 9, ky = r / 3, kx = r % 3;
            float v;
            if (f == 1) {
                v = gt[(((size_t)b * CC + c) * HH + (py * 3 + ky)) * WW + (px * 3 + kx)];
            } else {
                // bicubic (a=-0.75) downsample by f at fixed t=0.5, edge-clamped
                int oy = py * 3 + ky, ox = px * 3 + kx;
                int by = f * oy + (f >> 1) - 1;   // floor(src) along H
                int bx = f * ox + (f >> 1) - 1;   // floor(src) along W
                float acc = 0.f;
                #pragma unroll
                for (int i = 0; i < 4; i++) {
                    int yy = iclamp(by - 1 + i, 0, HH - 1);
                    const float* gr = gt + (((size_t)b * CC + c) * HH + yy) * WW;
                    float wy = cw[i];
                    #pragma unroll
                    for (int j = 0; j < 4; j++) {
                        int xx = iclamp(bx - 1 + j, 0, WW - 1);
                        acc += wy * cw[j] * gr[xx];
                    }
                }
                v = acc;
            }
            row[k] = v; nrm += v * v;
        }
        #pragma unroll
        for (int k = 27; k < DP; k++) row[k] = 0.f;
        qn[q] = nrm;
    }
}

// ---------------------------------------------------------------------------
// Kernel 2: fused dual fp32 WMMA GEMM (G1 = P1·Qᵀ, G2 = P2·Qᵀ with P2 = Q rows)
// + squared-distance epilogue + in-register argmin over all M columns.
// One wave32 owns a 16-row n-tile; streams 189 m-tiles, SOFTWARE-PIPELINED:
// the B-operand tile (7 x v2f) + qn for tile mt+1 is loaded before the WMMAs
// of tile mt so the 7-load clause overlaps 14 WMMAs + epilogue.
// ---------------------------------------------------------------------------
__global__ void __launch_bounds__(128) bbl_score_argmin(
    const float* __restrict__ P1, const float* __restrict__ Q,
    const float* __restrict__ n1, const float* __restrict__ qn,
    int* __restrict__ bestIdx)
{
    int lane  = threadIdx.x & 31;
    int wave  = blockIdx.x * (blockDim.x >> 5) + (threadIdx.x >> 5);
    int b     = wave / (NN / 16);
    int nt    = wave % (NN / 16);
    int lrow  = lane & 15;   // A-row / B-col within tile
    int khalf = lane >> 4;   // which K pair this lane holds

    const float* Pa    = P1 + ((size_t)(b * NN + nt * 16 + lrow)) * DP + 2 * khalf;
    const float* Qbase = Q + (size_t)b * MM * DP;
    const float* Bld   = Qbase + (size_t)lrow * DP + 2 * khalf;  // + mt*16*DP per tile
    const float* qnb   = qn + (size_t)b * MM;
    const float* Pb    = Qbase + (size_t)(nt * 16 + lrow) * DP + 2 * khalf; // P2 rows == Q rows

    // Preload both A operands (constant across the m loop): 7 K-steps of 4.
    v2f a1[7], a2[7];
    #pragma unroll
    for (int s = 0; s < 7; s++) {
        a1[s] = *(const v2f*)(Pa + 4 * s);
        a2[s] = *(const v2f*)(Pb + 4 * s);
    }

    // Per-lane n-row norms for the 8 accumulator rows this lane sees.
    float n1v[8], n2v[8];
    #pragma unroll
    for (int r = 0; r < 8; r++) {
        int row = nt * 16 + r + 8 * khalf;
        n1v[r] = n1[(size_t)b * NN + row];
        n2v[r] = qnb[row];   // ||p2_row||^2 == ||q_row||^2 for row < 2304
    }

    float bestS[8]; int bestM[8];
    #pragma unroll
    for (int r = 0; r < 8; r++) { bestS[r] = 3.0e38f; bestM[r] = 0; }

    // ---- software pipeline: prologue loads tile 0 ----
    v2f bb[7]; float qv;
    {
        const float* qrow = Bld;
        #pragma unroll
        for (int s = 0; s < 7; s++) bb[s] = *(const v2f*)(qrow + 4 * s);
        qv = qnb[lrow];
    }

    for (int mt = 0; mt < NT; mt++) {
        // Issue next tile's loads first so they overlap this tile's compute.
        v2f bbn[7]; float qvn;
        if (mt + 1 < NT) {
            const float* qrow = Bld + (size_t)(mt + 1) * 16 * DP;
            #pragma unroll
            for (int s = 0; s < 7; s++) bbn[s] = *(const v2f*)(qrow + 4 * s);
            qvn = qnb[(mt + 1) * 16 + lrow];
        } else {
            #pragma unroll
            for (int s = 0; s < 7; s++) bbn[s] = bb[s];
            qvn = qv;
        }

        v8f c1 = {0.f,0.f,0.f,0.f,0.f,0.f,0.f,0.f};
        v8f c2 = {0.f,0.f,0.f,0.f,0.f,0.f,0.f,0.f};
        #pragma unroll
        for (int s = 0; s < 7; s++) {
            c1 = __builtin_amdgcn_wmma_f32_16x16x4_f32(false, a1[s], false, bb[s],
                                                       (short)0, c1, false, false);
            c2 = __builtin_amdgcn_wmma_f32_16x16x4_f32(false, a2[s], false, bb[s],
                                                       (short)0, c2, false, false);
        }
        int mi = mt * 16 + lrow;
        #pragma unroll
        for (int r = 0; r < 8; r++) {
            float d1 = fmaxf(n1v[r] + qv - 2.0f * c1[r], 0.0f);
            float d2 = fmaxf(n2v[r] + qv - 2.0f * c2[r], 0.0f);
            float sc = d1 + d2;                       // 1/27 scale preserves argmin
            if (sc < bestS[r]) { bestS[r] = sc; bestM[r] = mi; }  // strict <: keep earliest m
        }

        #pragma unroll
        for (int s = 0; s < 7; s++) bb[s] = bbn[s];
        qv = qvn;
    }

    // Min-reduce (score, index) across the 16-lane group; tie -> smaller m.
    #pragma unroll
    for (int off = 1; off < 16; off <<= 1) {
        #pragma unroll
        for (int r = 0; r < 8; r++) {
            float os = __shfl_xor(bestS[r], off, 32);
            int   om = __shfl_xor(bestM[r], off, 32);
            if (os < bestS[r] || (os == bestS[r] && om < bestM[r])) {
                bestS[r] = os; bestM[r] = om;
            }
        }
    }
    if (lrow == 0) {
        #pragma unroll
        for (int r = 0; r < 8; r++)
            bestIdx[(size_t)b * NN + nt * 16 + r + 8 * khalf] = bestM[r];
    }
}

// ---------------------------------------------------------------------------
// Kernel 3: gather best-buddy row, L1 distance, block-level reduction.
// ---------------------------------------------------------------------------
__global__ void bbl_l1(const float* __restrict__ P1, const float* __restrict__ Q,
                       const int* __restrict__ bestIdx, float* __restrict__ partials)
{
    int t = blockIdx.x * blockDim.x + threadIdx.x;
    float s = 0.f;
    if (t < BB * NN) {
        int b = t / NN;
        int m = bestIdx[t];
        const float* pr = P1 + (size_t)t * DP;
        const float* qr = Q + ((size_t)b * MM + m) * DP;
        #pragma unroll
        for (int k = 0; k < 27; k++) s += fabsf(pr[k] - qr[k]);
    }
    __shared__ float red[256];
    red[threadIdx.x] = s;
    __syncthreads();
    for (int off = 128; off > 0; off >>= 1) {
        if (threadIdx.x < off) red[threadIdx.x] += red[threadIdx.x + off];
        __syncthreads();
    }
    if (threadIdx.x == 0) partials[blockIdx.x] = red[0];
}

// Kernel 4: deterministic final reduction -> mean.
__global__ void bbl_final(const float* __restrict__ partials, float* __restrict__ out,
                          int nPartials)
{
    __shared__ float red[256];
    float s = (threadIdx.x < nPartials) ? partials[threadIdx.x] : 0.f;
    red[threadIdx.x] = s;
    __syncthreads();
    for (int off = 128; off > 0; off >>= 1) {
        if (threadIdx.x < off) red[threadIdx.x] += red[threadIdx.x + off];
        __syncthreads();
    }
    if (threadIdx.x == 0)
        out[0] = red[0] / (float)(BB * NN * 27);   // 995328 elements
}

// ---------------------------------------------------------------------------
extern "C" void kernel_launch(void* const* d_in, const int* in_sizes, int n_in,
                              void* d_out, int out_size, void* d_ws, size_t ws_size,
                              hipStream_t stream)
{
    const float* x  = (const float*)d_in[0];
    const float* gt = (const float*)d_in[1];
    float* out = (float*)d_out;

    // Workspace layout (all offsets 256B-aligned)
    char* ws = (char*)d_ws;
    size_t off = 0;
    float* P1 = (float*)(ws + off); off += (size_t)BB * NN * DP * sizeof(float); // 4.72 MB
    float* Q  = (float*)(ws + off); off += (size_t)BB * MM * DP * sizeof(float); // 6.19 MB
    float* n1 = (float*)(ws + off); off += (size_t)BB * NN * sizeof(float);
    float* qn = (float*)(ws + off); off += (size_t)BB * MM * sizeof(float);
    int* bestIdx = (int*)(ws + off); off += (size_t)BB * NN * sizeof(int);
    float* partials = (float*)(ws + off); off += 256 * sizeof(float);
    (void)ws_size; (void)in_sizes; (void)n_in; (void)out_size;

    // 1) Build patch matrices + norms
    {
        int total = BB * (NN + MM);                 // 85248 rows
        int blocks = (total + 255) / 256;
        bbl_build<<<blocks, 256, 0, stream>>>(x, gt, P1, Q, n1, qn);
    }
    // 2) Fused WMMA scores + argmin: 16*(2304/16)=2304 waves, 4 waves/block
    {
        int waves = BB * (NN / 16);
        bbl_score_argmin<<<waves / 4, 128, 0, stream>>>(P1, Q, n1, qn, bestIdx);
    }
    // 3) Gather + L1 partial sums: 36864 threads -> 144 partials
    {
        int total = BB * NN;
        int blocks = (total + 255) / 256;           // 144
        bbl_l1<<<blocks, 256, 0, stream>>>(P1, Q, bestIdx, partials);
        // 4) Final reduction
        bbl_final<<<1, 256, 0, stream>>>(partials, out, blocks);
    }
}